// ChamferLoss_48447231099485
// MI455X (gfx1250) — compile-verified
//
#include <hip/hip_runtime.h>
#include <hip/hip_bf16.h>

typedef __attribute__((ext_vector_type(2))) float v2f;
typedef __attribute__((ext_vector_type(8))) float v8f;

#define BATCH        4
#define NPTS         8192     // N == M == 8192
#define MC           2048     // Q columns staged in LDS per chunk (32 KB)
#define WGSIZE       256      // 8 wave32 waves
#define ROWS_PER_WG  128      // 8 waves * 16 rows each

// One direction of the Chamfer loss:
//   for each point p in P, find min_q |p - q|^2, sum over the row strip.
// P = [B,3,N] queries, Q = [B,3,M] references.
// Each wave owns a 16-row strip of P; loops over all M columns of Q in
// LDS-staged chunks, one v_wmma_f32_16x16x4_f32 per 16x16 tile:
//   A row  = [p0, p1, p2, 1]
//   B col  = [-2q0, -2q1, -2q2, |q|^2]
//   C init = |p|^2  (broadcast per row)
//   D      = |p|^2 + |q|^2 - 2 p.q   (exact squared distance tile)
//
// LDS B tile is stored split by K-pair so each lane issues one ds_load_b64
// that is directly its B operand (no per-iteration selects):
//   plane 0: (-2q0, -2q1)  -> lanes 0..15  (K=0,1)
//   plane 1: (-2q2, |q|^2) -> lanes 16..31 (K=2,3)
__global__ __launch_bounds__(WGSIZE)
void chamfer_dir_kernel(const float* __restrict__ P,
                        const float* __restrict__ Q,
                        float* __restrict__ partials,  // one slot per block
                        int N, int M)
{
    __shared__ float2 q_s[2][MC];
    __shared__ float  wsum[16];

    const int b    = blockIdx.y;
    const int tid  = threadIdx.x;
    const int lane = tid & 31;
    const int wave = tid >> 5;
    const int half = lane >> 4;   // 0 -> K={0,1}, 1 -> K={2,3} (ISA A/B layout)
    const int l16  = lane & 15;

    const float* Px = P + (size_t)b * 3 * N;
    const float* Qx = Q + (size_t)b * 3 * M;

    // ---- A matrix (16x4 f32) for this wave's row strip ----
    const int n0   = (blockIdx.x * (WGSIZE / 32) + wave) * 16;
    const int rowA = n0 + l16;
    const float p0 = Px[0 * N + rowA];
    const float p1 = Px[1 * N + rowA];
    const float p2 = Px[2 * N + rowA];
    v2f a;
    a.x = half ? p2   : p0;   // K=0 or K=2
    a.y = half ? 1.0f : p1;   // K=1 or K=3 (ones column carries |q|^2)

    // ---- C matrix: |p|^2 broadcast per row (VGPR r -> rows r / r+8) ----
    v8f c;
#pragma unroll
    for (int r = 0; r < 8; ++r) {
        const int row = n0 + r + 8 * half;
        const float u0 = Px[0 * N + row];
        const float u1 = Px[1 * N + row];
        const float u2 = Px[2 * N + row];
        c[r] = u0 * u0 + u1 * u1 + u2 * u2;
    }

    // Per-lane-half plane base: hoisted out of the inner loop.
    const float2* __restrict__ qplane = &q_s[half][0];

    float rmin[8];
#pragma unroll
    for (int r = 0; r < 8; ++r) rmin[r] = 3.4e38f;

    for (int m0 = 0; m0 < M; m0 += MC) {
        __syncthreads();  // previous chunk fully consumed before overwrite
        // Stage MC columns of Q into the two K-pair planes.
        // 2048 cols / 256 threads = 8 uniform iterations (EXEC stays full).
        for (int i = tid; i < MC; i += WGSIZE) {
            const int col = m0 + i;
            const float q0 = Qx[0 * M + col];
            const float q1 = Qx[1 * M + col];
            const float q2 = Qx[2 * M + col];
            q_s[0][i] = make_float2(-2.0f * q0, -2.0f * q1);
            q_s[1][i] = make_float2(-2.0f * q2, q0 * q0 + q1 * q1 + q2 * q2);
        }
        // Prefetch next chunk toward L2/L0 (global_prefetch_b8 on gfx1250).
        if (m0 + MC < M) {
            const int col = m0 + MC + tid * (MC / WGSIZE);
            __builtin_prefetch(&Qx[0 * M + col], 0, 0);
            __builtin_prefetch(&Qx[1 * M + col], 0, 0);
            __builtin_prefetch(&Qx[2 * M + col], 0, 0);
        }
        __syncthreads();

        // 16 columns per WMMA step: one ds_load_b64 + one WMMA + 8 mins.
        for (int s = 0; s < MC / 16; ++s) {
            const float2 bq = qplane[s * 16 + l16];
            v2f bv;
            bv.x = bq.x;
            bv.y = bq.y;
            v8f d = __builtin_amdgcn_wmma_f32_16x16x4_f32(
                /*neg_a=*/false, a, /*neg_b=*/false, bv,
                /*c_mod=*/(short)0, c, /*reuse_a=*/false, /*reuse_b=*/false);
#pragma unroll
            for (int r = 0; r < 8; ++r) rmin[r] = fminf(rmin[r], d[r]);
        }
    }

    // Row-min across the 16 lanes of each half (rows split lo/hi per ISA).
#pragma unroll
    for (int r = 0; r < 8; ++r) {
#pragma unroll
        for (int mask = 8; mask >= 1; mask >>= 1)
            rmin[r] = fminf(rmin[r], __shfl_xor(rmin[r], mask, 32));
    }
    float s = 0.0f;
#pragma unroll
    for (int r = 0; r < 8; ++r) s += rmin[r];  // lane0: rows 0..7, lane16: rows 8..15

    if (l16 == 0) wsum[wave * 2 + half] = s;
    __syncthreads();
    if (tid == 0) {
        float t = 0.0f;
#pragma unroll
        for (int i = 0; i < 16; ++i) t += wsum[i];
        partials[blockIdx.y * gridDim.x + blockIdx.x] = t;  // unique slot: deterministic
    }
}

// Fixed-order reduction of per-block partials -> scalar loss.
__global__ void chamfer_finalize_kernel(const float* __restrict__ part,
                                        float* __restrict__ out,
                                        int nper, float invNB, float invMB)
{
    if (threadIdx.x == 0 && blockIdx.x == 0) {
        float sx = 0.0f, sy = 0.0f;
        for (int i = 0; i < nper; ++i) sx += part[i];
        for (int i = 0; i < nper; ++i) sy += part[nper + i];
        out[0] = sx * invNB + sy * invMB;
    }
}

extern "C" void kernel_launch(void* const* d_in, const int* in_sizes, int n_in,
                              void* d_out, int out_size, void* d_ws, size_t ws_size,
                              hipStream_t stream) {
    (void)in_sizes; (void)n_in; (void)out_size; (void)ws_size;
    const float* x = (const float*)d_in[0];  // [B,3,N]
    const float* y = (const float*)d_in[1];  // [B,3,M]
    float* out  = (float*)d_out;
    float* part = (float*)d_ws;

    const int B = BATCH, N = NPTS, M = NPTS;
    const int blocks_per_pass = (N / ROWS_PER_WG) * B;  // 64 * 4 = 256

    dim3 grid(N / ROWS_PER_WG, B);
    // Pass 1: for each x, min over y  -> partials[0 .. 255]
    chamfer_dir_kernel<<<grid, WGSIZE, 0, stream>>>(x, y, part, N, M);
    // Pass 2: for each y, min over x  -> partials[256 .. 511]
    chamfer_dir_kernel<<<grid, WGSIZE, 0, stream>>>(y, x, part + blocks_per_pass, M, N);

    chamfer_finalize_kernel<<<1, 32, 0, stream>>>(
        part, out, blocks_per_pass,
        1.0f / ((float)B * (float)N), 1.0f / ((float)B * (float)M));
}